// CoPEAttention_88390426952436
// MI455X (gfx1250) — compile-verified
//
#include <hip/hip_runtime.h>
#include <math.h>

// CoPE attention, CLS-row only.  Algebraically restructured:
//   q0 = emb[:,0,:]@Wq.T + bq                       (WMMA f32 16x16x4, padded M=16)
//   u  = q0@Wk, c = q0.bk, li = q0@pos_emb          (WMMA f32 16x16x4)
//   logits[b,t] = scale*(emb[b,t].u[b] + c[b])      (streaming, BW-bound)
//   CoPE suffix-scan + bias interp + softmax        (LDS scan per batch)
//   w[b] = sum_t attn[b,t]*emb[b,t,:]               (streaming, L2 hit)
//   out  = w@Wv.T + bv                              (WMMA f32 16x16x4)
// All M-padding lives in workspace buffers so WMMA fragment loads are
// unconditional global_load_b64 (no per-lane exec predication).

typedef float v2f __attribute__((ext_vector_type(2)));
typedef float v8f __attribute__((ext_vector_type(8)));

namespace {
constexpr int kB = 8;
constexpr int kS = 2048;
constexpr int kD = 768;
constexpr int kN = 512;
constexpr int kChunks = 4;                       // t-chunks for weighted sum
constexpr float kScale = 0.03608439182435161f;   // 1/sqrt(768)

// workspace offsets in floats
constexpr int A0_OFF = 0;        // 16*768  padded CLS embeddings
constexpr int Q0_OFF = 12288;    // 16*768  padded q0
constexpr int U_OFF  = 24576;    // 16*768  padded u
constexpr int LI_OFF = 36864;    // 16*512  padded li
constexpr int C_OFF  = 45056;    // 8
constexpr int LG_OFF = 45120;    // 8*2048  logits -> attn in place
constexpr int WP_OFF = 61504;    // 4*8*768 partial weighted sums
constexpr int W_OFF  = 86080;    // 16*768  padded w
} // namespace

__device__ __forceinline__ v2f ld2(const float* p) {
  return *reinterpret_cast<const v2f*>(p);
}

// stage CLS rows into padded [16 x 768] A buffer (rows 8..15 zero)
__global__ void k_stage(const float* __restrict__ emb, float* __restrict__ a0) {
  const int idx = blockIdx.x * 256 + threadIdx.x;  // 0..12287
  const int row = idx / kD, col = idx % kD;
  a0[idx] = (row < kB) ? emb[(size_t)row * kS * kD + col] : 0.0f;
}

// ---------------- WMMA GEMM kernels (one wave per 16-col tile) ----------------

__global__ void k_q0(const float* __restrict__ a0, const float* __restrict__ Wq,
                     const float* __restrict__ bq, float* __restrict__ q0) {
  const int lane = threadIdx.x;
  const int m = lane & 15;
  const bool hi = lane >= 16;
  const int col = blockIdx.x * 16 + (lane & 15);
  v8f acc = {};
#pragma unroll 4
  for (int k0 = 0; k0 < kD; k0 += 4) {
    const int ka = k0 + (hi ? 2 : 0);
    const v2f a = ld2(a0 + m * kD + ka);
    const v2f b = ld2(Wq + (size_t)col * kD + ka);   // B[k][n] = Wq[n][k]
    acc = __builtin_amdgcn_wmma_f32_16x16x4_f32(false, a, false, b, (short)0,
                                                acc, false, false);
  }
  const float bias = bq[col];
#pragma unroll
  for (int j = 0; j < 8; ++j) {
    const int mm = j + (hi ? 8 : 0);
    q0[mm * kD + col] = (mm < kB) ? (acc[j] + bias) : 0.0f;  // select, no branch
  }
}

// blocks 0..47: u = q0@Wk ; blocks 48..79: li = q0@pos_emb ; block 80: c = q0.bk
__global__ void k_u_li_c(const float* __restrict__ q0, const float* __restrict__ Wk,
                         const float* __restrict__ bk, const float* __restrict__ pe,
                         float* __restrict__ u, float* __restrict__ li,
                         float* __restrict__ cvec) {
  const int p = blockIdx.x;
  const int lane = threadIdx.x;
  if (p == 80) {
    if (lane < kB) {
      float acc = 0.f;
      for (int k = 0; k < kD; ++k) acc += q0[lane * kD + k] * bk[k];
      cvec[lane] = acc;
    }
    return;
  }
  const int m = lane & 15;
  const bool hi = lane >= 16;
  v8f acc = {};
  if (p < 48) {
    const int col = p * 16 + (lane & 15);
#pragma unroll 4
    for (int k0 = 0; k0 < kD; k0 += 4) {
      const int ka = k0 + (hi ? 2 : 0);
      const v2f a = ld2(q0 + m * kD + ka);
      v2f b;
      b.x = Wk[(size_t)ka * kD + col];               // B[k][n] = Wk[k][n]
      b.y = Wk[(size_t)(ka + 1) * kD + col];
      acc = __builtin_amdgcn_wmma_f32_16x16x4_f32(false, a, false, b, (short)0,
                                                  acc, false, false);
    }
#pragma unroll
    for (int j = 0; j < 8; ++j) {
      const int mm = j + (hi ? 8 : 0);
      u[mm * kD + col] = acc[j];                     // padded 16-row store
    }
  } else {
    const int col = (p - 48) * 16 + (lane & 15);
#pragma unroll 4
    for (int k0 = 0; k0 < kD; k0 += 4) {
      const int ka = k0 + (hi ? 2 : 0);
      const v2f a = ld2(q0 + m * kD + ka);
      v2f b;
      b.x = pe[(size_t)ka * kN + col];               // B[d][n] = pos_emb[d][n]
      b.y = pe[(size_t)(ka + 1) * kN + col];
      acc = __builtin_amdgcn_wmma_f32_16x16x4_f32(false, a, false, b, (short)0,
                                                  acc, false, false);
    }
#pragma unroll
    for (int j = 0; j < 8; ++j) {
      const int mm = j + (hi ? 8 : 0);
      li[mm * kN + col] = acc[j];
    }
  }
}

__global__ void k_out(const float* __restrict__ w, const float* __restrict__ Wv,
                      const float* __restrict__ bv, float* __restrict__ out) {
  const int lane = threadIdx.x;
  const int m = lane & 15;
  const bool hi = lane >= 16;
  const int col = blockIdx.x * 16 + (lane & 15);
  v8f acc = {};
#pragma unroll 4
  for (int k0 = 0; k0 < kD; k0 += 4) {
    const int ka = k0 + (hi ? 2 : 0);
    const v2f a = ld2(w + m * kD + ka);
    const v2f b = ld2(Wv + (size_t)col * kD + ka);   // B[e][d] = Wv[d][e]
    acc = __builtin_amdgcn_wmma_f32_16x16x4_f32(false, a, false, b, (short)0,
                                                acc, false, false);
  }
  const float bias = bv[col];
  if (!hi) {                                         // rows 0..7 only; uniform-ish
#pragma unroll
    for (int j = 0; j < 8; ++j) out[j * kD + col] = acc[j] + bias;
  }
}

// ---------------- streaming / scan kernels ----------------

// logits[b,t] = scale*(emb[b,t].u[b] + c[b]); one wave per row t
__global__ void k_logits(const float* __restrict__ emb, const int* __restrict__ mask,
                         const float* __restrict__ u, const float* __restrict__ cvec,
                         float* __restrict__ logits) {
  __shared__ float su[kD];
  const int b = blockIdx.y;
  const int tid = threadIdx.x;
  for (int j = tid; j < kD; j += 256) su[j] = u[b * kD + j];
  __syncthreads();
  const int wid = tid >> 5, lane = tid & 31;
  const int t = blockIdx.x * 8 + wid;
  const float* row = emb + ((size_t)b * kS + t) * kD;
  float acc = 0.f;
  for (int e = lane; e < kD; e += 32) acc += row[e] * su[e];
  for (int off = 16; off > 0; off >>= 1) acc += __shfl_xor(acc, off, 32);
  if (lane == 0) {
    float l = kScale * (acc + cvec[b]);
    if (mask[b * kS + t] == 0) l = -INFINITY;
    logits[b * kS + t] = l;
  }
}

__device__ inline float block_allreduce_max(float v, float* s_red) {
  const int lane = threadIdx.x & 31, wid = threadIdx.x >> 5;
  for (int off = 16; off > 0; off >>= 1) v = fmaxf(v, __shfl_xor(v, off, 32));
  if (lane == 0) s_red[wid] = v;
  __syncthreads();
  if (wid == 0) {
    float x = s_red[lane];
    for (int off = 16; off > 0; off >>= 1) x = fmaxf(x, __shfl_xor(x, off, 32));
    if (lane == 0) s_red[0] = x;
  }
  __syncthreads();
  const float r = s_red[0];
  __syncthreads();
  return r;
}

__device__ inline float block_allreduce_sum(float v, float* s_red) {
  const int lane = threadIdx.x & 31, wid = threadIdx.x >> 5;
  for (int off = 16; off > 0; off >>= 1) v += __shfl_xor(v, off, 32);
  if (lane == 0) s_red[wid] = v;
  __syncthreads();
  if (wid == 0) {
    float x = s_red[lane];
    for (int off = 16; off > 0; off >>= 1) x += __shfl_xor(x, off, 32);
    if (lane == 0) s_red[0] = x;
  }
  __syncthreads();
  const float r = s_red[0];
  __syncthreads();
  return r;
}

// gates=sigmoid(logits); suffix-scan; clamp; bias interp; softmax. 1 block / batch.
__global__ void k_cope_softmax(float* __restrict__ logits, const float* __restrict__ li) {
  __shared__ float s_pos[kS];
  __shared__ float s_red[32];
  const int b = blockIdx.x, tid = threadIdx.x;
  const int t0 = tid, t1 = tid + 1024;
  const float l0 = logits[b * kS + t0];
  const float l1 = logits[b * kS + t1];
  s_pos[t0] = 1.f / (1.f + expf(-l0));   // sigmoid(-inf)=0 handled
  s_pos[t1] = 1.f / (1.f + expf(-l1));
  __syncthreads();
  // Hillis-Steele suffix (reverse-cumsum) scan over 2048 elements
  for (int off = 1; off < kS; off <<= 1) {
    const float a0 = (t0 + off < kS) ? s_pos[t0 + off] : 0.f;
    const float a1 = (t1 + off < kS) ? s_pos[t1 + off] : 0.f;
    __syncthreads();
    s_pos[t0] += a0;
    s_pos[t1] += a1;
    __syncthreads();
  }
  const float* lib = li + b * kN;
  float nl[2];
  const float lv[2] = {l0, l1};
  const int tt[2] = {t0, t1};
#pragma unroll
  for (int q = 0; q < 2; ++q) {
    const float p = fminf(s_pos[tt[q]], (float)(kN - 1));
    const float pf = floorf(p);
    const int fi = (int)pf;
    const int ci = (int)ceilf(p);
    const float wfrac = p - pf;
    nl[q] = lv[q] + lib[ci] * wfrac + lib[fi] * (1.f - wfrac);
  }
  __syncthreads();
  const float mx = block_allreduce_max(fmaxf(nl[0], nl[1]), s_red);
  const float e0 = expf(nl[0] - mx);
  const float e1 = expf(nl[1] - mx);
  const float ssum = block_allreduce_sum(e0 + e1, s_red);
  const float inv = 1.f / ssum;
  logits[b * kS + t0] = e0 * inv;   // attn, in place
  logits[b * kS + t1] = e1 * inv;
}

// partial weighted sums over 512-row t-chunks; thread = channel e
__global__ void k_wsum(const float* __restrict__ emb, const float* __restrict__ attn,
                       float* __restrict__ wpart) {
  __shared__ float sa[kS / kChunks];
  const int b = blockIdx.y, ch = blockIdx.x;
  const int tid = threadIdx.x;
  const int tbase = ch * (kS / kChunks);
  if (tid < kS / kChunks) sa[tid] = attn[b * kS + tbase + tid];
  __syncthreads();
  float acc = 0.f;
  const float* base = emb + ((size_t)b * kS + tbase) * kD + tid;
  for (int i = 0; i < kS / kChunks; ++i) acc += sa[i] * base[(size_t)i * kD];
  wpart[(b * kChunks + ch) * kD + tid] = acc;
}

// reduce partials into padded [16 x 768] w (rows 8..15 zero)
__global__ void k_wreduce(const float* __restrict__ wpart, float* __restrict__ w) {
  const int idx = blockIdx.x * 256 + threadIdx.x;  // 0..12287
  const int b = idx / kD, e = idx % kD;
  float s = 0.f;
  if (b < kB) {
#pragma unroll
    for (int c = 0; c < kChunks; ++c) s += wpart[(b * kChunks + c) * kD + e];
  }
  w[idx] = s;
}

extern "C" void kernel_launch(void* const* d_in, const int* in_sizes, int n_in,
                              void* d_out, int out_size, void* d_ws, size_t ws_size,
                              hipStream_t stream) {
  (void)in_sizes; (void)n_in; (void)out_size; (void)ws_size;
  const float* emb  = (const float*)d_in[0];
  const int*   mask = (const int*)d_in[1];
  const float* Wq   = (const float*)d_in[2];
  const float* bq   = (const float*)d_in[3];
  const float* Wk   = (const float*)d_in[4];
  const float* bk   = (const float*)d_in[5];
  const float* Wv   = (const float*)d_in[6];
  const float* bv   = (const float*)d_in[7];
  const float* pe   = (const float*)d_in[8];
  float* out = (float*)d_out;
  float* ws  = (float*)d_ws;

  k_stage<<<dim3(48), dim3(256), 0, stream>>>(emb, ws + A0_OFF);
  k_q0<<<dim3(kD / 16), dim3(32), 0, stream>>>(ws + A0_OFF, Wq, bq, ws + Q0_OFF);
  k_u_li_c<<<dim3(48 + 32 + 1), dim3(32), 0, stream>>>(ws + Q0_OFF, Wk, bk, pe,
                                                       ws + U_OFF, ws + LI_OFF,
                                                       ws + C_OFF);
  k_logits<<<dim3(kS / 8, kB), dim3(256), 0, stream>>>(emb, mask, ws + U_OFF,
                                                       ws + C_OFF, ws + LG_OFF);
  k_cope_softmax<<<dim3(kB), dim3(1024), 0, stream>>>(ws + LG_OFF, ws + LI_OFF);
  k_wsum<<<dim3(kChunks, kB), dim3(768), 0, stream>>>(emb, ws + LG_OFF, ws + WP_OFF);
  k_wreduce<<<dim3(48), dim3(256), 0, stream>>>(ws + WP_OFF, ws + W_OFF);
  k_out<<<dim3(kD / 16), dim3(32), 0, stream>>>(ws + W_OFF, Wv, bv, out);
}